// WeightedPoissonBlendLoss_34935263986322
// MI455X (gfx1250) — compile-verified
//
#include <hip/hip_runtime.h>
#include <hip/hip_bf16.h>
#include <math.h>

typedef float f32x4  __attribute__((ext_vector_type(4)));
typedef float f32x4u __attribute__((ext_vector_type(4), aligned(4)));
typedef float f32x2  __attribute__((ext_vector_type(2)));
typedef float f32x8  __attribute__((ext_vector_type(8)));

constexpr int   H = 4096, W = 4096;
constexpr float WSCALE = 5.0f;
constexpr float EPSV   = 1e-5f;
constexpr int   BLK    = 256;
constexpr int   VEC    = 4;
constexpr int   NBLOCKS = (H * W) / (BLK * VEC); // 16384
constexpr int   RBLK   = 1024;                   // final-reduce block size

// ---------------------------------------------------------------------------
// Main fused elementwise pass: one thread handles 4 consecutive columns.
// Memory-bound: ~640 MB of traffic -> ~28 us floor at 23.3 TB/s.
// ---------------------------------------------------------------------------
__global__ __launch_bounds__(BLK) void wpbl_main(
    const float* __restrict__ target, const float* __restrict__ maskp,
    const float* __restrict__ srcp, float* __restrict__ out,
    float* __restrict__ partial)
{
  const int tid  = threadIdx.x;
  const int v    = blockIdx.x * BLK + tid;
  const int e    = v << 2;            // flat element index of first lane-elem
  const int i    = e / W;             // row
  const int j    = e & (W - 1);       // col (W power of two)
  const int base = e;                 // == i*W + j

  // ---- loads (b128 where possible; inputs have row-reuse -> keep RT) ----
  f32x4 t  = *(const f32x4*)(target + base);
  f32x4 sv = *(const f32x4*)(srcp   + base);
  f32x4 mv = *(const f32x4*)(maskp  + base);

  const bool has_right = (j + VEC) < W;       // false only for last vec of row
  float t_r = 0.0f, s_r = 0.0f, mk_r = 0.0f;
  if (has_right) {
    t_r  = target[base + VEC];
    s_r  = srcp[base + VEC];
    mk_r = maskp[base + VEC];
  }

  const bool has_down = (i + 1) < H;
  f32x4 t_d = {}, s_d = {}, m_d4 = {};
  f32x4 s_u = {}, m_u4 = {};
  if (has_down) {
    t_d  = *(const f32x4*)(target + base + W);
    s_d  = *(const f32x4*)(srcp   + base + W);
    m_d4 = *(const f32x4*)(maskp  + base + W);
  } else {
    // last row: diff_dup_last(mask, axis=0) needs row H-2
    s_u  = *(const f32x4*)(srcp  + base - W);
    m_u4 = *(const f32x4*)(maskp + base - W);
  }

  // ---- derived values (modified mask m = mask*(source!=0), sm = source*m) ----
  float tt[5], m[5], sm[5];
  float md[4], smd[4], mu[4], td[4];
#pragma unroll
  for (int k = 0; k < 4; ++k) {
    tt[k]  = t[k];
    m[k]   = (sv[k] != 0.0f) ? mv[k] : 0.0f;
    sm[k]  = sv[k] * m[k];
    td[k]  = t_d[k];
    md[k]  = (s_d[k] != 0.0f) ? m_d4[k] : 0.0f;
    smd[k] = s_d[k] * md[k];
    mu[k]  = (s_u[k] != 0.0f) ? m_u4[k] : 0.0f;
  }
  tt[4] = t_r;
  m[4]  = (s_r != 0.0f) ? mk_r : 0.0f;
  sm[4] = s_r * m[4];

  f32x4u wxv, wyv, gbxv, gbyv, gsxv, gsyv, zzv;
  float acc = 0.0f;
#pragma unroll
  for (int k = 0; k < 4; ++k) {
    const bool xvalid = (k < 3) || has_right;          // (j+k) < W-1
    float gbx = xvalid ? (tt[k + 1] - tt[k]) : 0.0f;   // diff append-last -> 0 at edge
    float gsx = xvalid ? (sm[k + 1] - sm[k]) : 0.0f;
    float gmx = (k == 3 && !has_right) ? (m[3] - m[2]) // diff dup-last at col W-1
                                       : (m[k + 1] - m[k]);

    float gby = has_down ? (td[k]  - tt[k]) : 0.0f;
    float gsy = has_down ? (smd[k] - sm[k]) : 0.0f;
    float gmy = has_down ? (md[k] - m[k]) : (m[k] - mu[k]); // dup-last at row H-1

    if ((gmx != 0.0f) || (gmy != 0.0f)) { gbx = 0.0f; gby = 0.0f; }

    const float inv_d = 1.0f / (sm[k] + EPSV);
    const float wx = m[k] * __expf(-WSCALE * fabsf(gsx * inv_d));
    const float wy = m[k] * __expf(-WSCALE * fabsf(gsy * inv_d));

    const float dx = wx * (gbx - gsx);
    const float dy = wy * (gby - gsy);
    acc += dx * dx + dy * dy;

    wxv[k] = wx;  wyv[k] = wy;
    gbxv[k] = gbx; gbyv[k] = gby;
    gsxv[k] = gsx; gsyv[k] = gsy;
    zzv[k]  = 0.0f;
  }

  // ---- streaming stores: outputs are never re-read on device -> TH=NT so the
  //      470 MB write stream doesn't evict the reusable input rows from L2 ----
  const int wrow  = 1 + i * (3 * W);
  __builtin_nontemporal_store(wxv, (f32x4u*)(out + wrow + j));
  __builtin_nontemporal_store(wyv, (f32x4u*)(out + wrow + W + j));
  __builtin_nontemporal_store(zzv, (f32x4u*)(out + wrow + 2 * W + j));
  const int gbase = 1 + H * 3 * W;
  __builtin_nontemporal_store(gbxv, (f32x4u*)(out + gbase + base));
  __builtin_nontemporal_store(gbyv, (f32x4u*)(out + gbase + H * W + base));
  __builtin_nontemporal_store(gsxv, (f32x4u*)(out + gbase + 2 * H * W + base));
  __builtin_nontemporal_store(gsyv, (f32x4u*)(out + gbase + 3 * H * W + base));

  // ---- deterministic block reduction of squared-error sum ----
  __shared__ float red[BLK];
  red[tid] = acc;
  __syncthreads();
#pragma unroll
  for (int s = BLK / 2; s > 0; s >>= 1) {
    if (tid < s) red[tid] += red[tid + s];
    __syncthreads();
  }
  if (tid == 0) partial[blockIdx.x] = red[0];
}

// ---------------------------------------------------------------------------
// Final reduction: 1 block x 1024 threads (32 waves).
//  * each thread issues 4 independent b128 loads (all in flight -> not
//    serialized like a single-wave scalar loop)
//  * LDS tree (fixed order, deterministic) down to 32 partials
//  * ALL waves broadcast those 32 values into lanes and run the WMMA
//    ones-reduction with EXEC all-1s (WMMA requires full EXEC per wave):
//      A[m][0]=s[m], A[m][2]=s[m+16] (f32 16x4 A layout), B=ones
//      => D[m][n] = s[m]+s[m+16]; col(lane) = c0+..+c7; total = col + col^16
// ---------------------------------------------------------------------------
__global__ __launch_bounds__(RBLK) void wpbl_reduce(
    const float* __restrict__ partial, int n, float* __restrict__ out)
{
  const int tid = threadIdx.x;
  const int nchunks = n >> 2;               // 4096 f32x4 chunks

  float s = 0.0f;
#pragma unroll
  for (int q = 0; q < 4; ++q) {             // 4096 / 1024 = 4 chunks/thread
    f32x4 p = ((const f32x4*)partial)[tid + q * RBLK];
    s += p[0]; s += p[1]; s += p[2]; s += p[3];
  }
  (void)nchunks;

  __shared__ float red[RBLK];
  red[tid] = s;
  __syncthreads();
#pragma unroll
  for (int st = RBLK / 2; st >= 32; st >>= 1) {
    if (tid < st) red[tid] += red[tid + st];
    __syncthreads();
  }

  // broadcast the surviving 32 partials into every wave's lanes
  float lanev = red[tid & 31];

  f32x2 a; a.x = lanev; a.y = 0.0f;
  f32x2 b; b.x = 1.0f;  b.y = 1.0f;
  f32x8 c = {};
  c = __builtin_amdgcn_wmma_f32_16x16x4_f32(
      /*neg_a=*/false, a, /*neg_b=*/false, b,
      /*c_mod=*/(short)0, c, /*reuse_a=*/false, /*reuse_b=*/false);

  float col = c[0] + c[1] + c[2] + c[3] + c[4] + c[5] + c[6] + c[7];
  float other = __shfl(col, (int)((tid ^ 16) & 31), 32);
  float total = col + other;
  if (tid == 0) {
    out[0] = total * (1.0f / 16777216.0f);  // mean over H*W (exact pow2 scale)
  }
}

extern "C" void kernel_launch(void* const* d_in, const int* in_sizes, int n_in,
                              void* d_out, int out_size, void* d_ws, size_t ws_size,
                              hipStream_t stream) {
  // setup_inputs() order: target, mask, source (all fp32, H*W each)
  const float* target = (const float*)d_in[0];
  const float* mask   = (const float*)d_in[1];
  const float* source = (const float*)d_in[2];
  float* out     = (float*)d_out;
  float* partial = (float*)d_ws;   // NBLOCKS floats = 64 KB scratch

  wpbl_main<<<NBLOCKS, BLK, 0, stream>>>(target, mask, source, out, partial);
  wpbl_reduce<<<1, RBLK, 0, stream>>>(partial, NBLOCKS, out);
}